// SoftNNLossEuclidean_44160853738086
// MI455X (gfx1250) — compile-verified
//
#include <hip/hip_runtime.h>
#include <hip/hip_bf16.h>

typedef float v2f __attribute__((ext_vector_type(2)));
typedef float v8f __attribute__((ext_vector_type(8)));

#define D 128
#define COLCHUNK 512

static __host__ __device__ inline int cdiv(int a, int b) { return (a + b - 1) / b; }

// ---------------------------------------------------------------------------
// Stage 0: zero the per-row accumulators and the scalar output.
// ---------------------------------------------------------------------------
__global__ void zero_accum_kernel(float* __restrict__ posSum,
                                  float* __restrict__ denomSum,
                                  float* __restrict__ out, int n) {
    int i = blockIdx.x * blockDim.x + threadIdx.x;
    if (i < n) { posSum[i] = 0.0f; denomSum[i] = 0.0f; }
    if (i == 0) out[0] = 0.0f;
}

// ---------------------------------------------------------------------------
// Stage 1: squared L2 norm of each length-128 row. One thread per row,
// 32x float4 loads -> global_load_b128 stream.
// ---------------------------------------------------------------------------
__global__ void rownorm_kernel(const float* __restrict__ A,
                               float* __restrict__ out, int rows) {
    int i = blockIdx.x * blockDim.x + threadIdx.x;
    if (i >= rows) return;
    const float4* p = (const float4*)(A + (size_t)i * D);
    float s = 0.0f;
#pragma unroll
    for (int t = 0; t < D / 4; ++t) {
        float4 v = p[t];
        s = fmaf(v.x, v.x, s); s = fmaf(v.y, v.y, s);
        s = fmaf(v.z, v.z, s); s = fmaf(v.w, v.w, s);
    }
    out[i] = s;
}

// ---------------------------------------------------------------------------
// Stage 2: fused  rowAccum[i] += sum_j exp(-alpha * ||x_i - y_j||)
// One wave handles 16 rows x 16 cols tiles with V_WMMA_F32_16X16X4_F32.
//   grid.x = N/16 (row blocks), grid.y = cols/COLCHUNK, block = 256 (8 waves)
// Wave w inside a block walks column tiles  chunkStart + w*16, step 8*16.
//
// A-frag (16x4 f32): lane L (m=L&15, hi=L>>4) holds X[rowBase+m][k + 2*hi + {0,1}]
// B-frag (4x16 f32): lane L holds Y[jBase+m][k + 2*hi + {0,1}]   (B = Y^T)
// C/D    (16x16 f32): lane L, VGPR r  ->  element (M = r + 8*hi, N = L&15)
//
// B tile is batch-loaded into a register array before the WMMA chain so the
// 32 global_load_b64 issue as one clause and their latency overlaps the
// 32-deep WMMA dependency chain (instead of load->wait->wmma lockstep).
// ---------------------------------------------------------------------------
__global__ __launch_bounds__(256)
void simsum_kernel(const float* __restrict__ X, const float* __restrict__ x2,
                   const float* __restrict__ Y, const float* __restrict__ y2,
                   int cols, float alpha, float* __restrict__ rowAccum) {
    const int lane = threadIdx.x & 31;
    const int wave = threadIdx.x >> 5;
    const int m    = lane & 15;
    const int hi   = lane >> 4;

    const int rowBase    = blockIdx.x * 16;
    const int chunkStart = blockIdx.y * COLCHUNK;
    const int chunkEnd   = min(chunkStart + COLCHUNK, cols);

    // Preload the whole 16x128 A tile for this wave: 32 k-steps of float2.
    v2f a[D / 4];
    {
        const float* xrow = X + (size_t)(rowBase + m) * D + 2 * hi;
#pragma unroll
        for (int ks = 0; ks < D / 4; ++ks)
            a[ks] = *(const v2f*)(xrow + ks * 4);
    }
    // ||x||^2 for the 8 rows this lane's C registers map to.
    float xn[8];
#pragma unroll
    for (int r = 0; r < 8; ++r) xn[r] = x2[rowBase + r + 8 * hi];

    float acc[8];
#pragma unroll
    for (int r = 0; r < 8; ++r) acc[r] = 0.0f;

    // Wave-uniform column-tile loop: EXEC stays all-ones around the WMMAs.
    for (int j = chunkStart + wave * 16; j < chunkEnd; j += 8 * 16) {
        const float* yrow = Y + (size_t)(j + m) * D + 2 * hi;

        // Prefetch next tile's B rows (wave-uniform guard; gfx1250
        // global_prefetch_b8 — fetches the full cacheline per lane).
        if (j + 8 * 16 < chunkEnd)
            __builtin_prefetch(Y + (size_t)(j + 8 * 16 + m) * D + 2 * hi, 0, 3);

        // Batch-load the full B tile (32 k-steps) into registers first.
        v2f b[D / 4];
#pragma unroll
        for (int ks = 0; ks < D / 4; ++ks)
            b[ks] = *(const v2f*)(yrow + ks * 4);

        v8f c = {0.0f, 0.0f, 0.0f, 0.0f, 0.0f, 0.0f, 0.0f, 0.0f};
#pragma unroll
        for (int ks = 0; ks < D / 4; ++ks) {
            // D = A(16x4) * B(4x16) + C ; full f32 matrix pipe
            c = __builtin_amdgcn_wmma_f32_16x16x4_f32(
                    /*neg_a=*/false, a[ks], /*neg_b=*/false, b[ks],
                    /*c_mod=*/(short)0, c, /*reuse_a=*/false, /*reuse_b=*/false);
        }

        const float yn = y2[j + m];
#pragma unroll
        for (int r = 0; r < 8; ++r) {
            float sq = fmaf(-2.0f, c[r], xn[r] + yn);
            sq = fmaxf(sq, 0.0f);
            float dist = __builtin_amdgcn_sqrtf(sq);  // raw v_sqrt_f32
            acc[r] += __expf(-alpha * dist);          // v_exp_f32 path
        }
    }

    // Reduce each acc[r] across the 16-lane half-wave (rows r and r+8),
    // then one atomicAdd per row from lanes 0 and 16.
#pragma unroll
    for (int r = 0; r < 8; ++r) {
        float v = acc[r];
        v += __shfl_xor(v, 1, 32);
        v += __shfl_xor(v, 2, 32);
        v += __shfl_xor(v, 4, 32);
        v += __shfl_xor(v, 8, 32);
        if (m == 0)
            atomicAdd(&rowAccum[rowBase + r + 8 * hi], v);
    }
}

// ---------------------------------------------------------------------------
// Stage 3: loss = sum_i [ log(denom_i) - log(pos_i) ]
// ---------------------------------------------------------------------------
__global__ void loss_reduce_kernel(const float* __restrict__ posSum,
                                   const float* __restrict__ denomSum,
                                   float* __restrict__ out, int n) {
    __shared__ float sm[256];
    const int tid = threadIdx.x;
    float s = 0.0f;
    for (int i = blockIdx.x * blockDim.x + tid; i < n; i += gridDim.x * blockDim.x)
        s += logf(denomSum[i]) - logf(posSum[i]);
    sm[tid] = s;
    __syncthreads();
    for (int o = 128; o > 0; o >>= 1) {
        if (tid < o) sm[tid] += sm[tid + o];
        __syncthreads();
    }
    if (tid == 0) atomicAdd(out, sm[0]);
}

// ---------------------------------------------------------------------------
extern "C" void kernel_launch(void* const* d_in, const int* in_sizes, int n_in,
                              void* d_out, int out_size, void* d_ws, size_t ws_size,
                              hipStream_t stream) {
    const float* X = (const float*)d_in[0];  // image_views      [N,128]
    const float* P = (const float*)d_in[1];  // positive_views   [P,128]
    const float* Q = (const float*)d_in[2];  // negative_views   [Q,128]
    const float* O = (const float*)d_in[3];  // other_embeddings [M,128]
    const int N  = in_sizes[0] / D;
    const int Pn = in_sizes[1] / D;
    const int Qn = in_sizes[2] / D;
    const int Mn = in_sizes[3] / D;

    float* ws = (float*)d_ws;
    float* x2 = ws;            // N
    float* p2 = x2 + N;        // Pn
    float* q2 = p2 + Pn;       // Qn
    float* o2 = q2 + Qn;       // Mn
    float* posSum   = o2 + Mn;       // N
    float* denomSum = posSum + N;    // N
    float* out = (float*)d_out;

    // Stage 1: row norms (and Stage 0 zeroing — independent, same stream).
    rownorm_kernel<<<cdiv(N,  256), 256, 0, stream>>>(X, x2, N);
    rownorm_kernel<<<cdiv(Pn, 256), 256, 0, stream>>>(P, p2, Pn);
    rownorm_kernel<<<cdiv(Qn, 256), 256, 0, stream>>>(Q, q2, Qn);
    rownorm_kernel<<<cdiv(Mn, 256), 256, 0, stream>>>(O, o2, Mn);
    zero_accum_kernel<<<cdiv(N, 256), 256, 0, stream>>>(posSum, denomSum, out, N);

    // Stage 2: fused WMMA distance + exp row-sums.
    // pos/neg: exp(-W*d/T) = exp(-4d).  others: exp(-d/T) = exp(-2d).
    {
        dim3 g(N / 16, cdiv(Pn, COLCHUNK));
        simsum_kernel<<<g, 256, 0, stream>>>(X, x2, P, p2, Pn, 4.0f, posSum);
    }
    {
        dim3 g(N / 16, cdiv(Qn, COLCHUNK));
        simsum_kernel<<<g, 256, 0, stream>>>(X, x2, Q, q2, Qn, 4.0f, denomSum);
    }
    {
        dim3 g(N / 16, cdiv(Mn, COLCHUNK));
        simsum_kernel<<<g, 256, 0, stream>>>(X, x2, O, o2, Mn, 2.0f, denomSum);
    }

    // Stage 3: scalar loss.
    loss_reduce_kernel<<<16, 256, 0, stream>>>(posSum, denomSum, out, N);
}